// PositionHeuristicSearcher_45569603011118
// MI455X (gfx1250) — compile-verified
//
#include <hip/hip_runtime.h>
#include <stdint.h>
#include <math.h>

typedef __attribute__((ext_vector_type(2))) float v2f;
typedef __attribute__((ext_vector_type(8))) float v8f;

#define NEGV (-1e9f)

constexpr int Bb = 4, Qd = 2048, Td = 2048, Dd = 512;

// ---------------------------------------------------------------------------
// 1) Row reciprocal norms: rnorm[row] = 1/||x[row,:]||  (one block per row)
// ---------------------------------------------------------------------------
__global__ void rownorm_kernel(const float* __restrict__ x,
                               float* __restrict__ rnorm, int d) {
  const int row = blockIdx.x;
  const int tid = threadIdx.x;                 // 128 threads
  const float* p = x + (size_t)row * d;
  float s = 0.f;
  for (int i = tid; i < d; i += blockDim.x) { float v = p[i]; s += v * v; }
  for (int off = 16; off > 0; off >>= 1) s += __shfl_xor(s, off, 32);
  __shared__ float wsum[4];
  const int lane = tid & 31, w = tid >> 5;
  if (lane == 0) wsum[w] = s;
  __syncthreads();
  if (tid == 0) {
    float t = 0.f;
    for (int i = 0; i < (int)(blockDim.x >> 5); ++i) t += wsum[i];
    rnorm[row] = 1.0f / sqrtf(t);
  }
}

// ---------------------------------------------------------------------------
// 2) Batched normalized GEMM: logits[b,q,t] = <dec_n[b,q,:], tgt_n[b,t,:]>
//    128x128 tile / block (8 waves), fp32 WMMA 16x16x4, K streamed via LDS.
// ---------------------------------------------------------------------------
#define LDSP 18   // row pad (even -> 8B aligned float2 ds loads, no 4-way conflicts)

__global__ __launch_bounds__(256) void gemm_logits_kernel(
    const float* __restrict__ A, const float* __restrict__ Bm,
    const float* __restrict__ rA, const float* __restrict__ rB,
    float* __restrict__ C) {
  __shared__ float As[128][LDSP];
  __shared__ float Bs[128][LDSP];

  const int b  = blockIdx.z;
  const int m0 = blockIdx.y * 128;
  const int n0 = blockIdx.x * 128;
  const float* Ab  = A  + (size_t)b * Qd * Dd;
  const float* Bbp = Bm + (size_t)b * Td * Dd;
  const float* rAb = rA + (size_t)b * Qd;
  const float* rBb = rB + (size_t)b * Td;
  float* Cb = C + (size_t)b * Qd * Td;

  const int tid  = threadIdx.x;
  const int lane = tid & 31;
  const int w    = tid >> 5;
  const int wm   = (w & 1) * 64;   // wave M offset within block tile
  const int wn   = (w >> 1) * 32;  // wave N offset within block tile
  const int l16  = lane & 15;
  const int hi   = lane >> 4;

  v8f acc[4][2];
  for (int i = 0; i < 4; ++i)
    for (int j = 0; j < 2; ++j) acc[i][j] = (v8f){};

  for (int kc = 0; kc < Dd; kc += 16) {
    // Fill LDS: 128 rows x 16 cols of A and B-tile, normalization fused in.
    for (int i = tid; i < 512; i += 256) {
      const int row = i >> 2;
      const int cg  = (i & 3) << 2;
      const float4 va = *(const float4*)(Ab + (size_t)(m0 + row) * Dd + kc + cg);
      const float ra = rAb[m0 + row];
      As[row][cg + 0] = va.x * ra; As[row][cg + 1] = va.y * ra;
      As[row][cg + 2] = va.z * ra; As[row][cg + 3] = va.w * ra;
      const float4 vb = *(const float4*)(Bbp + (size_t)(n0 + row) * Dd + kc + cg);
      const float rb = rBb[n0 + row];
      Bs[row][cg + 0] = vb.x * rb; Bs[row][cg + 1] = vb.y * rb;
      Bs[row][cg + 2] = vb.z * rb; Bs[row][cg + 3] = vb.w * rb;
    }
    __syncthreads();

    for (int kk = 0; kk < 16; kk += 4) {
      const int ka = kk + 2 * hi;
      v2f afrag[4], bfrag[2];
      for (int mt = 0; mt < 4; ++mt) {
        const float* p = &As[wm + mt * 16 + l16][ka];
        v2f a; a.x = p[0]; a.y = p[1]; afrag[mt] = a;
      }
      for (int nt = 0; nt < 2; ++nt) {
        const float* p = &Bs[wn + nt * 16 + l16][ka];
        v2f bf; bf.x = p[0]; bf.y = p[1]; bfrag[nt] = bf;
      }
      for (int mt = 0; mt < 4; ++mt)
        for (int nt = 0; nt < 2; ++nt)
          acc[mt][nt] = __builtin_amdgcn_wmma_f32_16x16x4_f32(
              false, afrag[mt], false, bfrag[nt],
              (short)0, acc[mt][nt], false, false);
    }
    __syncthreads();
  }

  // Store: VGPR r of 16x16 f32 C holds row (r + 8*hi), col = lane&15.
  for (int mt = 0; mt < 4; ++mt)
    for (int nt = 0; nt < 2; ++nt) {
      const int col = n0 + wn + nt * 16 + l16;
      for (int r = 0; r < 8; ++r) {
        const int row = m0 + wm + mt * 16 + r + 8 * hi;
        Cb[(size_t)row * Td + col] = acc[mt][nt][r];
      }
    }
}

// ---------------------------------------------------------------------------
// 3) Zero one_hot + reset iteration counter
// ---------------------------------------------------------------------------
__global__ void zero_onehot_kernel(float* __restrict__ onehot, size_t n,
                                   int* __restrict__ iter_counter) {
  const size_t i = (size_t)blockIdx.x * blockDim.x + threadIdx.x;
  if (i < n) onehot[i] = 0.f;
  if (i == 0) *iter_counter = 0;
}

// ---------------------------------------------------------------------------
// 4) Greedy max-connect matching. One persistent block per batch element.
//    All state in LDS; column claims via 64-bit ds_max atomics with key
//    (orderedFloat(val) << 32) | (0x7FFFFFFF - row)  -> ties go to lowest row,
//    matching jnp.argmax semantics exactly.
// ---------------------------------------------------------------------------
__global__ __launch_bounds__(1024) void match_kernel(
    const float* __restrict__ logits,
    const uint8_t* __restrict__ mask_dec, const uint8_t* __restrict__ mask_tgt,
    int* __restrict__ index_out, float* __restrict__ onehot,
    int* __restrict__ iter_counter) {
  __shared__ uint8_t row_done[Qd];
  __shared__ uint8_t col_done[Td];
  __shared__ uint8_t assigned_s[Qd];
  __shared__ uint8_t claiming_s[Qd];
  __shared__ int     idx_s[Qd];
  __shared__ int     bcol_s[Qd];
  __shared__ unsigned long long claim_s[Td];
  __shared__ int     remaining;

  const int b = blockIdx.x;
  const float* Lb = logits + (size_t)b * Qd * Td;
  const int tid = threadIdx.x;
  const int nthr = blockDim.x;           // 1024
  const int lane = tid & 31, wv = tid >> 5, nwv = nthr >> 5;

  for (int q = tid; q < Qd; q += nthr) {
    row_done[q] = mask_dec[b * Qd + q] ? 0 : 1;
    assigned_s[q] = 0; claiming_s[q] = 0; idx_s[q] = 0; bcol_s[q] = 0;
  }
  for (int t = tid; t < Td; t += nthr) {
    col_done[t] = mask_tgt[b * Td + t] ? 0 : 1;
    claim_s[t] = 0ULL;
  }
  __syncthreads();

  int it = 0;
  for (; it < Qd; ++it) {
    // any(~row_done)?
    if (tid == 0) remaining = 0;
    __syncthreads();
    int cnt = 0;
    for (int q = tid; q < Qd; q += nthr) cnt += row_done[q] ? 0 : 1;
    for (int off = 16; off > 0; off >>= 1) cnt += __shfl_xor(cnt, off, 32);
    if (lane == 0 && cnt) atomicAdd(&remaining, cnt);
    __syncthreads();
    if (remaining == 0) break;

    // Phase A: per-row argmax over available columns (one wave per row).
    for (int q = wv; q < Qd; q += nwv) {
      if (row_done[q]) continue;
      const float* rowp = Lb + (size_t)q * Td;
      float bv = NEGV; int bc = 0x7FFFFFFF;
      for (int t = lane; t < Td; t += 32) {
        const float v = col_done[t] ? NEGV : rowp[t];
        if (v > bv) { bv = v; bc = t; }      // ascending t: first max wins
      }
      for (int off = 16; off > 0; off >>= 1) {
        const float ov = __shfl_xor(bv, off, 32);
        const int   oc = __shfl_xor(bc, off, 32);
        if (ov > bv || (ov == bv && oc < bc)) { bv = ov; bc = oc; }
      }
      if (lane == 0) {
        bcol_s[q] = bc;
        const bool claim = (bv > NEGV * 0.5f);
        claiming_s[q] = claim ? 1 : 0;
        if (claim) {
          unsigned int ob = __float_as_uint(bv);
          ob = (ob & 0x80000000u) ? ~ob : (ob | 0x80000000u);  // order-preserving
          const unsigned long long key =
              ((unsigned long long)ob << 32) | (unsigned int)(0x7FFFFFFF - q);
          atomicMax(&claim_s[bc], key);
        } else {
          row_done[q] = 1;  // stuck row retires
        }
      }
    }
    __syncthreads();

    // Phase C: resolve winners.
    for (int q = tid; q < Qd; q += nthr) {
      if (!claiming_s[q]) continue;
      claiming_s[q] = 0;
      const int bc = bcol_s[q];
      const unsigned long long key = claim_s[bc];
      const int win = 0x7FFFFFFF - (int)(unsigned int)(key & 0xFFFFFFFFu);
      if (win == q) {
        idx_s[q] = bc;
        assigned_s[q] = 1;
        row_done[q] = 1;
        col_done[bc] = 1;     // unique winner -> no race
      }
    }
    __syncthreads();
    for (int t = tid; t < Td; t += nthr) claim_s[t] = 0ULL;
    __syncthreads();
  }

  // Outputs
  for (int q = tid; q < Qd; q += nthr) {
    index_out[b * Qd + q] = idx_s[q];
    if (assigned_s[q])
      onehot[(size_t)b * Qd * Td + (size_t)q * Td + idx_s[q]] = 1.0f;
  }
  if (tid == 0) atomicMax(iter_counter, it);
}

__global__ void write_iters_kernel(const int* __restrict__ c,
                                   int* __restrict__ out) { *out = *c; }

// ---------------------------------------------------------------------------
// Launch
// ---------------------------------------------------------------------------
extern "C" void kernel_launch(void* const* d_in, const int* in_sizes, int n_in,
                              void* d_out, int out_size, void* d_ws, size_t ws_size,
                              hipStream_t stream) {
  const float*   dec  = (const float*)d_in[0];
  const float*   tgt  = (const float*)d_in[1];
  const uint8_t* mdec = (const uint8_t*)d_in[2];
  const uint8_t* mtgt = (const uint8_t*)d_in[3];

  float* out       = (float*)d_out;
  float* logits    = out;                                       // B*Q*T f32
  int*   index_out = (int*)(out + (size_t)Bb * Qd * Td);        // B*Q   i32
  float* onehot    = out + (size_t)Bb * Qd * Td + Bb * Qd;      // B*Q*T f32
  int*   iters_out = (int*)(out + 2 * (size_t)Bb * Qd * Td + Bb * Qd); // 1 i32

  float* rA      = (float*)d_ws;          // B*Q recip norms
  float* rB      = rA + Bb * Qd;          // B*T recip norms
  int*   counter = (int*)(rB + Bb * Td);  // iteration counter

  rownorm_kernel<<<Bb * Qd, 128, 0, stream>>>(dec, rA, Dd);
  rownorm_kernel<<<Bb * Td, 128, 0, stream>>>(tgt, rB, Dd);

  dim3 gg(Td / 128, Qd / 128, Bb);
  gemm_logits_kernel<<<gg, 256, 0, stream>>>(dec, tgt, rA, rB, logits);

  const size_t n = (size_t)Bb * Qd * Td;
  zero_onehot_kernel<<<(unsigned)((n + 255) / 256), 256, 0, stream>>>(onehot, n, counter);

  match_kernel<<<Bb, 1024, 0, stream>>>(logits, mdec, mtgt, index_out, onehot, counter);
  write_iters_kernel<<<1, 1, 0, stream>>>(counter, iters_out);
}